// GIN_74680891343606
// MI455X (gfx1250) — compile-verified
//
#include <hip/hip_runtime.h>

typedef float v2f __attribute__((ext_vector_type(2)));
typedef float v8f __attribute__((ext_vector_type(8)));

#define DH 128

// ---------------------------------------------------------------------------
// Zero the aggregation buffer (float4 granularity).
// ---------------------------------------------------------------------------
__global__ void gin_zero_f4(float4* __restrict__ p, int n4) {
    int i = blockIdx.x * blockDim.x + threadIdx.x;
    if (i < n4) p[i] = make_float4(0.f, 0.f, 0.f, 0.f);
}

// ---------------------------------------------------------------------------
// Edge scatter-add: one wave (32 lanes) per edge, float4 per lane (128 dims).
// agg[dst[e]][:] += h[src[e]][:]
// ---------------------------------------------------------------------------
__global__ void gin_scatter(const float* __restrict__ h,
                            const int* __restrict__ src,
                            const int* __restrict__ dst,
                            float* __restrict__ agg,
                            int n_edges) {
    const int e    = blockIdx.x * 8 + (threadIdx.x >> 5);  // 8 waves per block
    const int lane = threadIdx.x & 31;
    if (e >= n_edges) return;

    const int s = src[e];
    const int d = dst[e];

    const float4 v = *(const float4*)(h + (size_t)s * DH + lane * 4);
    float* out = agg + (size_t)d * DH + lane * 4;
    // hardware global_atomic_add_f32 (no CAS loop)
    unsafeAtomicAdd(out + 0, v.x);
    unsafeAtomicAdd(out + 1, v.y);
    unsafeAtomicAdd(out + 2, v.z);
    unsafeAtomicAdd(out + 3, v.w);
}

// ---------------------------------------------------------------------------
// Fused GIN layer GEMM:  hout = (hin + agg) @ W + bias
//   - block = 256 threads = 8 waves, owns a 16-row tile of nodes
//   - wave w computes output columns [16w, 16w+16)
//   - K loop: 32 x v_wmma_f32_16x16x4_f32 (exact fp32)
// Fragment layouts per CDNA5 ISA 7.12.2:
//   A 16x4 f32: lanes 0-15 row M=lane, VGPR{0,1}=K{0,1}; lanes 16-31 K{2,3}
//   B 4x16 f32: lanes 0-15 cols, VGPR0 rows {0|2}, VGPR1 rows {1|3}
//   C/D 16x16 f32: VGPR r -> row r (+8 for lanes 16-31), col = lane%16
// ---------------------------------------------------------------------------
__global__ void gin_gemm(const float* __restrict__ hin,
                         const float* __restrict__ agg,
                         const float* __restrict__ W,
                         const float* __restrict__ bias,
                         float* __restrict__ hout) {
    // 16 rows x 128 cols, padded to 132 floats/row: bank offset 4 per row ->
    // 16 rows map to distinct banks for column-strided A-fragment reads.
    __shared__ float xs[16][DH + 4];

    const int rowBase = blockIdx.x * 16;
    const int tid     = threadIdx.x;

    // ---- stage x = hin + agg into LDS (8 floats per thread) ----
    {
        const int idx = tid * 8;          // 0..2040, 256*8 = 2048 = 16*128
        const int r   = idx >> 7;         // /128
        const int c   = idx & (DH - 1);   // %128
        const size_t g = (size_t)(rowBase + r) * DH + c;
        const float4 h0 = *(const float4*)(hin + g);
        const float4 h1 = *(const float4*)(hin + g + 4);
        const float4 a0 = *(const float4*)(agg + g);
        const float4 a1 = *(const float4*)(agg + g + 4);
        *(float4*)(&xs[r][c])     = make_float4(h0.x + a0.x, h0.y + a0.y,
                                                h0.z + a0.z, h0.w + a0.w);
        *(float4*)(&xs[r][c + 4]) = make_float4(h1.x + a1.x, h1.y + a1.y,
                                                h1.z + a1.z, h1.w + a1.w);
    }
    __syncthreads();

    const int wave    = tid >> 5;         // 0..7
    const int lane    = tid & 31;
    const int colBase = wave * 16;
    const int n       = lane & 15;        // col within tile / row M for A
    const int kOff    = (lane < 16) ? 0 : 2;

    v8f acc = {};                         // C = 0
    #pragma unroll 4
    for (int k4 = 0; k4 < DH; k4 += 4) {
        // A fragment from LDS (row = n, K = k4+kOff, k4+kOff+1) : 8B aligned
        v2f a = *(const v2f*)(&xs[n][k4 + kOff]);
        // B fragment from global W (row-major [128][128])
        v2f b;
        b.x = W[(size_t)(k4 + kOff)     * DH + colBase + n];
        b.y = W[(size_t)(k4 + kOff + 1) * DH + colBase + n];
        acc = __builtin_amdgcn_wmma_f32_16x16x4_f32(
                  /*neg_a=*/false, a, /*neg_b=*/false, b,
                  /*c_mod=*/(short)0, acc, /*reuse_a=*/false, /*reuse_b=*/false);
    }

    // ---- epilogue: bias add + store ----
    const int   mHi = (lane < 16) ? 0 : 8;
    const float bv  = bias[colBase + n];
    #pragma unroll
    for (int r = 0; r < 8; ++r) {
        hout[(size_t)(rowBase + r + mHi) * DH + colBase + n] = acc[r] + bv;
    }
}

// ---------------------------------------------------------------------------
// Host-side launch: 4 GIN layers, ping-pong through workspace.
// d_in order: h, edge_index, W0,b0, W1,b1, W2,b2, W3,b3
// ---------------------------------------------------------------------------
extern "C" void kernel_launch(void* const* d_in, const int* in_sizes, int n_in,
                              void* d_out, int out_size, void* d_ws, size_t ws_size,
                              hipStream_t stream) {
    const float* h   = (const float*)d_in[0];
    const int*   ei  = (const int*)d_in[1];
    const int    n_nodes = in_sizes[0] / DH;   // 20000
    const int    n_edges = in_sizes[1] / 2;    // 640000
    const int*   src = ei;
    const int*   dst = ei + n_edges;

    const float* Wm[4] = { (const float*)d_in[2], (const float*)d_in[4],
                           (const float*)d_in[6], (const float*)d_in[8] };
    const float* Bm[4] = { (const float*)d_in[3], (const float*)d_in[5],
                           (const float*)d_in[7], (const float*)d_in[9] };

    float* out = (float*)d_out;
    float* agg = (float*)d_ws;
    float* hA  = agg + (size_t)n_nodes * DH;
    float* hB  = hA  + (size_t)n_nodes * DH;

    const int n4        = n_nodes * DH / 4;            // float4 count of agg
    const int zeroGrid  = (n4 + 255) / 256;
    const int scatGrid  = (n_edges + 7) / 8;           // 8 edges (waves) / block
    const int gemmGrid  = n_nodes / 16;                // 20000 % 16 == 0

    const float* cur = h;
    for (int l = 0; l < 4; ++l) {
        gin_zero_f4<<<zeroGrid, 256, 0, stream>>>((float4*)agg, n4);
        gin_scatter<<<scatGrid, 256, 0, stream>>>(cur, src, dst, agg, n_edges);
        float* nxt = (l == 3) ? out : ((l & 1) ? hB : hA);
        gin_gemm<<<gemmGrid, 256, 0, stream>>>(cur, agg, Wm[l], Bm[l], nxt);
        cur = nxt;
    }
}